// Conv_12060268167596
// MI455X (gfx1250) — compile-verified
//
#include <hip/hip_runtime.h>

typedef float v8f __attribute__((ext_vector_type(8)));
typedef float v2f __attribute__((ext_vector_type(2)));

#define IMG_H 512
#define IMG_W 512
#define PATCH_ROWS 20
#define PATCH_COLS 24   // padded row stride (bytes: 96, multiple of 8 -> b64-aligned)
#define WAVES_PER_BLOCK 4

// One wave computes one 16x16 output tile:
//   out[y,x] = sum_dy sum_c patch[y+dy, c] * w[dy, c-x]   (banded B, 5 taps/column)
// => 25 x V_WMMA_F32_16X16X4_F32 accumulating into a single 16x16 f32 C.
__global__ __launch_bounds__(128) void conv5x5_wmma_kernel(
    const float* __restrict__ X,
    const float* __restrict__ Kw,
    float* __restrict__ Out)
{
    // B fragments: [chunk t][lane][vgpr pair] -> float2 per lane per chunk
    __shared__ __align__(16) float sB[25 * 32 * 2];
    // per-wave 20x20 input patch, row stride 24 floats
    __shared__ __align__(16) float sPatch[WAVES_PER_BLOCK][PATCH_ROWS * PATCH_COLS];

    const int tid  = threadIdx.x;
    const int lane = tid & 31;
    const int wid  = tid >> 5;

    // ---- build banded-weight B fragments once per block (weight-only, tile-invariant)
    // WMMA B 4x16 layout: VGPR v, lanes 0-15 -> K = v, lanes 16-31 -> K = v + 2; N = lane&15
    for (int e = tid; e < 25 * 64; e += blockDim.x) {
        int t    = e >> 6;         // chunk 0..24  (t = dy*5 + ci)
        int l    = (e >> 1) & 31;  // lane
        int v    = e & 1;          // which of the float2 pair
        int dy   = t / 5;
        int ci   = t - dy * 5;
        int half = l >> 4;
        int n    = l & 15;
        int c    = ci * 4 + 2 * half + v;   // K column within 0..19
        int dx   = c - n;
        sB[e] = (dx >= 0 && dx <= 4) ? Kw[dy * 5 + dx] : 0.0f;
    }
    __syncthreads();

    // ---- which tile does this wave own
    const int tile = blockIdx.x * WAVES_PER_BLOCK + wid;      // 0 .. 131071
    const int img  = tile >> 10;                              // / (32*32)
    const int rem  = tile & 1023;
    const int y0   = (rem >> 5) << 4;
    const int x0   = (rem & 31) << 4;
    const float* src = X + (long)img * (IMG_H * IMG_W);

    // ---- stage 20x20 zero-padded patch into LDS (13 iters x 32 lanes covers 400 elems)
    float* patch = sPatch[wid];
#pragma unroll
    for (int i = 0; i < 13; ++i) {
        int e = i * 32 + lane;
        if (e < PATCH_ROWS * 20) {
            int r  = e / 20;
            int cc = e - r * 20;
            int gy = y0 + r - 2;
            int gx = x0 + cc - 2;
            float val = 0.0f;
            if (gy >= 0 && gy < IMG_H && gx >= 0 && gx < IMG_W)
                val = src[gy * IMG_W + gx];
            patch[r * PATCH_COLS + cc] = val;
        }
    }
    __syncthreads();

    const int m    = lane & 15;   // output row within tile (A-matrix M)
    const int half = lane >> 4;   // K half-select

    // keep all 25 B fragments resident in VGPRs for the tile
    v2f b[25];
#pragma unroll
    for (int t = 0; t < 25; ++t)
        b[t] = *(const v2f*)&sB[(t * 32 + lane) * 2];

    v8f acc = {};
#pragma unroll
    for (int dy = 0; dy < 5; ++dy) {
#pragma unroll
        for (int ci = 0; ci < 5; ++ci) {
            const int t = dy * 5 + ci;
            // A 16x4 layout: lane holds A[m, 2*half + {0,1}] -> one ds_load_b64
            v2f a = *(const v2f*)&patch[(m + dy) * PATCH_COLS + ci * 4 + 2 * half];
            acc = __builtin_amdgcn_wmma_f32_16x16x4_f32(
                false, a, false, b[t], (short)0, acc, false, false);
        }
    }

    // ---- store 16x16 tile: C/D layout VGPR r -> M = r + 8*half, N = lane&15
    float* dst = Out + (long)img * (IMG_H * IMG_W);
#pragma unroll
    for (int r = 0; r < 8; ++r) {
        dst[(long)(y0 + r + 8 * half) * IMG_W + (x0 + m)] = acc[r];
    }
}

extern "C" void kernel_launch(void* const* d_in, const int* in_sizes, int n_in,
                              void* d_out, int out_size, void* d_ws, size_t ws_size,
                              hipStream_t stream) {
    const float* X  = (const float*)d_in[0];   // (4,32,512,512) fp32
    const float* Kw = (const float*)d_in[1];   // (5,5) fp32
    // d_in[2]=stride(1), d_in[3]=padding(2) -- fixed by the reference setup
    float* Out = (float*)d_out;

    const int n_tiles = 4 * 32 * (IMG_H / 16) * (IMG_W / 16); // 131072
    const int blocks  = n_tiles / WAVES_PER_BLOCK;            // 32768
    conv5x5_wmma_kernel<<<blocks, 128, 0, stream>>>(X, Kw, Out);
}